// RGCN_76613626626428
// MI455X (gfx1250) — compile-verified
//
#include <hip/hip_runtime.h>

typedef __attribute__((ext_vector_type(2))) float v2f;
typedef __attribute__((ext_vector_type(8))) float v8f;

#define N_UE 50000
#define N_AP 5000
#define NE   1600000

__device__ __forceinline__ float sigmoidf_(float x) {
    return 1.0f / (1.0f + __expf(-x));
}

__global__ void fill_zero_kernel(float* __restrict__ p, int n) {
    int i = blockIdx.x * blockDim.x + threadIdx.x;
    if (i < n) p[i] = 0.0f;
}

// --------------------------------------------------------------------------
// Edge MLP on uplink edges, via chained V_WMMA_F32_16X16X4_F32.
// One wave = 16 edges. A (16x4 f32): lanes 0-15 hold (K=0,1), lanes 16-31
// hold (K=2,3). tmp row = [xu0, xu1, xa0, xa1, ev0, ev1, 0, 0]; two WMMAs
// cover K=0..3 and K=4..7. C init = ba1[n] broadcast (bias is per-N = lane).
// Second layer 16->1 done with a 16-lane xor-butterfly per half.
// --------------------------------------------------------------------------
__global__ __launch_bounds__(256) void edge_update_wmma(
    const float* __restrict__ x_ue, const float* __restrict__ x_ap,
    float* __restrict__ ea_up,
    const int* __restrict__ src, const int* __restrict__ dst,
    const float* __restrict__ Wa1, const float* __restrict__ ba1,
    const float* __restrict__ Wa2, const float* __restrict__ ba2)
{
    const int lane = threadIdx.x & 31;
    const int n    = lane & 15;      // N index (out feature) / row within half
    const int half = lane >> 4;      // 0: K=0..1 (and K=4..5), 1: K=2..3 (6..7)
    const int wid  = threadIdx.x >> 5;
    const int base = (blockIdx.x * 8 + wid) * 16;
    const int e    = base + n;       // edge handled by this lane's A row

    // A operand, WMMA #1: half0 -> UE features of edge e, half1 -> AP features
    int idx = half ? dst[e] : src[e];
    const float* xp = half ? (x_ap + 2 * idx) : (x_ue + 2 * idx);
    float a0 = xp[0], a1 = xp[1];

    // Edge features (K=4,5), only half0 contributes; half1 is zero padding
    float ev0 = ea_up[2 * e], ev1 = ea_up[2 * e + 1];

    v2f A1; A1[0] = a0; A1[1] = a1;
    v2f B1; B1[0] = Wa1[(half * 2) * 16 + n];
            B1[1] = Wa1[(half * 2 + 1) * 16 + n];
    v2f A2; A2[0] = half ? 0.0f : ev0;
            A2[1] = half ? 0.0f : ev1;
    v2f B2; B2[0] = half ? 0.0f : Wa1[4 * 16 + n];
            B2[1] = half ? 0.0f : Wa1[5 * 16 + n];

    float bias = ba1[n];
    v8f c;
    #pragma unroll
    for (int i = 0; i < 8; ++i) c[i] = bias;

    // h_pre = tmp @ Wa1 + ba1   (two chained 16x16x4 f32 WMMAs)
    c = __builtin_amdgcn_wmma_f32_16x16x4_f32(false, A1, false, B1,
                                              (short)0, c, false, false);
    c = __builtin_amdgcn_wmma_f32_16x16x4_f32(false, A2, false, B2,
                                              (short)0, c, false, false);

    // o = sigmoid(relu(h) @ Wa2 + ba2) ; reduce N across each 16-lane half
    float wa2n = Wa2[n];
    float b2   = ba2[0];
    float o[8];
    #pragma unroll
    for (int v = 0; v < 8; ++v) {
        float t = fmaxf(c[v], 0.0f) * wa2n;
        t += __shfl_xor(t, 1);
        t += __shfl_xor(t, 2);
        t += __shfl_xor(t, 4);
        t += __shfl_xor(t, 8);
        o[v] = sigmoidf_(t + b2);
    }

    // D rows: half0 VGPR v -> row v (0..7); half1 VGPR v -> row v+8.
    // Lanes n<8 of each half write rows half*8 + n.
    if (n < 8) {
        int row = half * 8 + n;
        float val = o[0];
        #pragma unroll
        for (int j = 1; j < 8; ++j) val = (n == j) ? o[j] : val;
        ea_up[2 * (base + row) + 1] = val;  // keep channel 0, replace channel 1
    }
}

// --------------------------------------------------------------------------
// Uplink messages: relu(x_ue[src]@Wn_ue+bn) + relu(ea_up@We_up+be),
// scatter-added (mean numerator + count) into AP accumulators.
// --------------------------------------------------------------------------
__global__ __launch_bounds__(256) void msg_up_kernel(
    const float* __restrict__ x_ue, const float* __restrict__ ea_up,
    const int* __restrict__ src, const int* __restrict__ dst,
    const float* __restrict__ Wn, const float* __restrict__ bn,
    const float* __restrict__ We, const float* __restrict__ be,
    float* __restrict__ sum_ap, float* __restrict__ cnt_ap)
{
    int e = blockIdx.x * 256 + threadIdx.x;
    int s = src[e], d = dst[e];
    float x0 = x_ue[2 * s], x1 = x_ue[2 * s + 1];
    float e0 = ea_up[2 * e], e1 = ea_up[2 * e + 1];
    float* sp = sum_ap + d * 32;
    #pragma unroll
    for (int j = 0; j < 32; ++j) {
        float m = fmaxf(fmaf(x1, Wn[32 + j], fmaf(x0, Wn[j], bn[j])), 0.0f)
                + fmaxf(fmaf(e1, We[32 + j], fmaf(e0, We[j], be[j])), 0.0f);
        atomicAdd(sp + j, m);
    }
    atomicAdd(cnt_ap + d, 1.0f);
}

// --------------------------------------------------------------------------
// Downlink messages: relu(ea_dn@We_dn+be) (+ relu(x_ap[src]@Wn_ap+bn) if l>0)
// --------------------------------------------------------------------------
__global__ __launch_bounds__(256) void msg_dn_kernel(
    const float* __restrict__ x_ap, const float* __restrict__ ea_dn,
    const int* __restrict__ src, const int* __restrict__ dst,
    const float* __restrict__ Wn, const float* __restrict__ bn,
    const float* __restrict__ We, const float* __restrict__ be,
    float* __restrict__ sum_ue, float* __restrict__ cnt_ue, int use_node)
{
    int e = blockIdx.x * 256 + threadIdx.x;
    int s = src[e], d = dst[e];
    float e0 = ea_dn[2 * e], e1 = ea_dn[2 * e + 1];
    float x0 = 0.0f, x1 = 0.0f;
    if (use_node) { x0 = x_ap[2 * s]; x1 = x_ap[2 * s + 1]; }
    float* sp = sum_ue + d * 32;
    #pragma unroll
    for (int j = 0; j < 32; ++j) {
        float m = fmaxf(fmaf(e1, We[32 + j], fmaf(e0, We[j], be[j])), 0.0f);
        if (use_node)
            m += fmaxf(fmaf(x1, Wn[32 + j], fmaf(x0, Wn[j], bn[j])), 0.0f);
        atomicAdd(sp + j, m);
    }
    atomicAdd(cnt_ue + d, 1.0f);
}

// --------------------------------------------------------------------------
// Node update: aggr = sum/max(cnt,1); t=[x, aggr+relu(x@Wn+bn)];
// p = sigmoid(relu(t@Wp1+bp1)@Wp2+bp2); x_new = [x0, p].
// --------------------------------------------------------------------------
__global__ __launch_bounds__(256) void node_update_kernel(
    const float* __restrict__ x, const float* __restrict__ sum,
    const float* __restrict__ cnt,
    const float* __restrict__ Wn, const float* __restrict__ bn,
    const float* __restrict__ Wp1, const float* __restrict__ bp1,
    const float* __restrict__ Wp2, const float* __restrict__ bp2,
    float* __restrict__ xnew, int n)
{
    int i = blockIdx.x * blockDim.x + threadIdx.x;
    if (i >= n) return;
    float x0 = x[2 * i], x1 = x[2 * i + 1];
    float cdiv = fmaxf(cnt[i], 1.0f);
    float t[34];
    t[0] = x0; t[1] = x1;
    #pragma unroll
    for (int j = 0; j < 32; ++j) {
        float aggr = sum[i * 32 + j] / cdiv;
        float res  = fmaxf(fmaf(x1, Wn[32 + j], fmaf(x0, Wn[j], bn[j])), 0.0f);
        t[2 + j] = aggr + res;
    }
    float acc2 = bp2[0];
    #pragma unroll
    for (int k = 0; k < 16; ++k) {
        float a = bp1[k];
        #pragma unroll
        for (int j = 0; j < 34; ++j) a = fmaf(t[j], Wp1[j * 16 + k], a);
        acc2 = fmaf(fmaxf(a, 0.0f), Wp2[k], acc2);
    }
    xnew[2 * i]     = x0;
    xnew[2 * i + 1] = sigmoidf_(acc2);
}

extern "C" void kernel_launch(void* const* d_in, const int* in_sizes, int n_in,
                              void* d_out, int out_size, void* d_ws, size_t ws_size,
                              hipStream_t stream) {
    const float* x_ue_in  = (const float*)d_in[0];
    const float* x_ap_in  = (const float*)d_in[1];
    const float* ea_up_in = (const float*)d_in[2];
    const float* ea_dn    = (const float*)d_in[3];
    const float* Wn_ue = (const float*)d_in[4];
    const float* bn_ue = (const float*)d_in[5];
    const float* Wn_ap = (const float*)d_in[6];
    const float* bn_ap = (const float*)d_in[7];
    const float* We_up = (const float*)d_in[8];
    const float* be_up = (const float*)d_in[9];
    const float* We_dn = (const float*)d_in[10];
    const float* be_dn = (const float*)d_in[11];
    const float* Wp1   = (const float*)d_in[12];
    const float* bp1   = (const float*)d_in[13];
    const float* Wp2   = (const float*)d_in[14];
    const float* bp2   = (const float*)d_in[15];
    const float* Wa1   = (const float*)d_in[16];
    const float* ba1   = (const float*)d_in[17];
    const float* Wa2   = (const float*)d_in[18];
    const float* ba2   = (const float*)d_in[19];
    const int* ei_up_src = (const int*)d_in[20];
    const int* ei_up_dst = (const int*)d_in[21];
    const int* ei_dn_src = (const int*)d_in[22];
    const int* ei_dn_dst = (const int*)d_in[23];

    // Workspace layout (floats): ~21 MB total
    float* ws     = (float*)d_ws;
    float* ea_up  = ws;                      // 2*NE
    float* x_ue_a = ea_up + 2 * NE;          // 2*N_UE
    float* x_ue_b = x_ue_a + 2 * N_UE;       // 2*N_UE
    float* x_ap_a = x_ue_b + 2 * N_UE;       // 2*N_AP
    float* x_ap_b = x_ap_a + 2 * N_AP;       // 2*N_AP
    float* sums   = x_ap_b + 2 * N_AP;       // contiguous zero-fill region:
    float* sum_ap = sums;                    //   32*N_AP
    float* cnt_ap = sum_ap + 32 * N_AP;      //   N_AP
    float* sum_ue = cnt_ap + N_AP;           //   32*N_UE
    float* cnt_ue = sum_ue + 32 * N_UE;      //   N_UE
    const int fill_n = 32 * N_AP + N_AP + 32 * N_UE + N_UE;

    hipMemcpyAsync(ea_up,  ea_up_in, (size_t)2 * NE   * sizeof(float), hipMemcpyDeviceToDevice, stream);
    hipMemcpyAsync(x_ue_a, x_ue_in,  (size_t)2 * N_UE * sizeof(float), hipMemcpyDeviceToDevice, stream);
    hipMemcpyAsync(x_ap_a, x_ap_in,  (size_t)2 * N_AP * sizeof(float), hipMemcpyDeviceToDevice, stream);

    float* xu_cur = x_ue_a; float* xu_new = x_ue_b;
    float* xa_cur = x_ap_a; float* xa_new = x_ap_b;

    for (int l = 0; l < 2; ++l) {
        fill_zero_kernel<<<(fill_n + 255) / 256, 256, 0, stream>>>(sums, fill_n);

        // edge_update on uplink edges (WMMA)
        edge_update_wmma<<<NE / 128, 256, 0, stream>>>(
            xu_cur, xa_cur, ea_up, ei_up_src, ei_up_dst,
            Wa1 + l * 96, ba1 + l * 16, Wa2 + l * 16, ba2 + l);

        // uplink: ue -> ap
        msg_up_kernel<<<NE / 256, 256, 0, stream>>>(
            xu_cur, ea_up, ei_up_src, ei_up_dst,
            Wn_ue + l * 64, bn_ue + l * 32, We_up + l * 64, be_up + l * 32,
            sum_ap, cnt_ap);
        node_update_kernel<<<(N_AP + 255) / 256, 256, 0, stream>>>(
            xa_cur, sum_ap, cnt_ap,
            Wn_ap + l * 64, bn_ap + l * 32,
            Wp1 + l * 544, bp1 + l * 16, Wp2 + l * 16, bp2 + l,
            xa_new, N_AP);

        // downlink: ap -> ue (uses OLD x_ap; node term only for l>0)
        msg_dn_kernel<<<NE / 256, 256, 0, stream>>>(
            xa_cur, ea_dn, ei_dn_src, ei_dn_dst,
            Wn_ap + l * 64, bn_ap + l * 32, We_dn + l * 64, be_dn + l * 32,
            sum_ue, cnt_ue, l > 0);
        node_update_kernel<<<(N_UE + 255) / 256, 256, 0, stream>>>(
            xu_cur, sum_ue, cnt_ue,
            Wn_ue + l * 64, bn_ue + l * 32,
            Wp1 + l * 544, bp1 + l * 16, Wp2 + l * 16, bp2 + l,
            xu_new, N_UE);

        float* t;
        t = xu_cur; xu_cur = xu_new; xu_new = t;
        t = xa_cur; xa_cur = xa_new; xa_new = t;
    }

    // Outputs: (x_ue, x_ap, ea_up, ea_dn) concatenated flat
    float* out = (float*)d_out;
    hipMemcpyAsync(out,                             xu_cur, (size_t)2 * N_UE * sizeof(float), hipMemcpyDeviceToDevice, stream);
    hipMemcpyAsync(out + 2 * N_UE,                  xa_cur, (size_t)2 * N_AP * sizeof(float), hipMemcpyDeviceToDevice, stream);
    hipMemcpyAsync(out + 2 * N_UE + 2 * N_AP,       ea_up,  (size_t)2 * NE   * sizeof(float), hipMemcpyDeviceToDevice, stream);
    hipMemcpyAsync(out + 2 * N_UE + 2 * N_AP + 2 * NE, ea_dn, (size_t)2 * NE * sizeof(float), hipMemcpyDeviceToDevice, stream);
}